// ScaledDotProductAttention_36790689858274
// MI455X (gfx1250) — compile-verified
//
#include <hip/hip_runtime.h>
#include <math.h>

typedef float v2f __attribute__((ext_vector_type(2)));
typedef float v8f __attribute__((ext_vector_type(8)));

#define BATCH 64
#define SEQ   1024
#define DIM   512

// workspace layout (float offsets)
#define WS_XS 0                       // (64,512)  column sums of x
#define WS_KS (WS_XS + BATCH*DIM)     // (64,512)  KS = XSUM*Wk^T + N*bk
#define WS_V  (WS_KS + BATCH*DIM)     // (64,512)  V  = KS*Wq
#define WS_C  (WS_V  + BATCH*DIM)     // (64,)     c[b] = bq . KS[b]
#define WS_S  (WS_C  + BATCH)         // (64,1024) mean scores

// ---------------------------------------------------------------- init -----
__global__ void init_zero_kernel(float* ws, float* out) {
    int idx = blockIdx.x * blockDim.x + threadIdx.x;   // 0..32767
    if (idx < BATCH * DIM) {
        ws[WS_XS + idx] = 0.0f;     // xsum accumulator
        out[idx]        = 0.0f;     // aggregated accumulator
    }
}

// ------------------------------------------------- pass 1: column sums -----
// grid: 64 batches x 8 chunks of 128 rows; 256 threads (d and d+256)
__global__ void colsum_kernel(const float* __restrict__ x, float* __restrict__ ws) {
    int b = blockIdx.x >> 3;
    int chunk = blockIdx.x & 7;
    const float* xb = x + (size_t)b * SEQ * DIM + (size_t)chunk * 128 * DIM;
    int d0 = threadIdx.x, d1 = threadIdx.x + 256;
    float s0 = 0.0f, s1 = 0.0f;
    #pragma unroll 4
    for (int n = 0; n < 128; ++n) {
        s0 += xb[n * DIM + d0];
        s1 += xb[n * DIM + d1];
    }
    atomicAdd(&ws[WS_XS + b * DIM + d0], s0);
    atomicAdd(&ws[WS_XS + b * DIM + d1], s1);
}

// --------------------------------- WMMA GEMM 1: KS = XSUM * Wk^T + N*bk ----
// D(64x512) = A(64x512) * B(512x512), B[k][n] = Wk[n*512 + k]
// one wave per 16x16 tile; 4 m-tiles * 32 n-tiles = 128 tiles
__global__ void gemm_wk_kernel(const float* __restrict__ Wk,
                               const float* __restrict__ bk,
                               float* __restrict__ ws) {
    int wave = threadIdx.x >> 5;
    int lane = threadIdx.x & 31;
    int tile = blockIdx.x * 4 + wave;
    int tm = tile >> 5;        // 0..3
    int tn = tile & 31;        // 0..31
    int r  = lane & 15;
    int h  = lane >> 4;

    const float* A = ws + WS_XS + (size_t)(tm * 16 + r) * DIM;  // row m of XSUM
    const float* Bp = Wk + (size_t)(tn * 16 + r) * DIM;         // B[k][n] = Wk[n][k]

    v8f acc = {};
    for (int k0 = 0; k0 < DIM; k0 += 4) {
        int k = k0 + 2 * h;
        v2f a = *(const v2f*)(A + k);
        v2f bfrag = *(const v2f*)(Bp + k);
        acc = __builtin_amdgcn_wmma_f32_16x16x4_f32(
            false, a, false, bfrag, (short)0, acc, false, false);
    }
    float bias = (float)SEQ * bk[tn * 16 + r];
    #pragma unroll
    for (int j = 0; j < 8; ++j) {
        int m = tm * 16 + j + 8 * h;
        ws[WS_KS + m * DIM + tn * 16 + r] = acc[j] + bias;
    }
}

// --------------------------------------------- c[b] = bq . KS[b] -----------
__global__ void dotc_kernel(const float* __restrict__ bq, float* __restrict__ ws) {
    int b = blockIdx.x;
    int lane = threadIdx.x;
    float s = 0.0f;
    for (int e = lane; e < DIM; e += 32)
        s += bq[e] * ws[WS_KS + b * DIM + e];
    for (int off = 16; off > 0; off >>= 1)
        s += __shfl_down(s, off, 32);
    if (lane == 0) ws[WS_C + b] = s;
}

// --------------------------------- WMMA GEMM 2: V = KS * Wq ----------------
// D(64x512) = A(64x512) * B(512x512), B[k][n] = Wq[k*512 + n]
__global__ void gemm_wq_kernel(const float* __restrict__ Wq, float* __restrict__ ws) {
    int wave = threadIdx.x >> 5;
    int lane = threadIdx.x & 31;
    int tile = blockIdx.x * 4 + wave;
    int tm = tile >> 5;
    int tn = tile & 31;
    int r  = lane & 15;
    int h  = lane >> 4;

    const float* A = ws + WS_KS + (size_t)(tm * 16 + r) * DIM;
    const float* Bc = Wq + (tn * 16 + r);   // column n; element k at Bc[k*512]

    v8f acc = {};
    for (int k0 = 0; k0 < DIM; k0 += 4) {
        int k = k0 + 2 * h;
        v2f a = *(const v2f*)(A + k);
        v2f bfrag;
        bfrag.x = Bc[(size_t)k * DIM];
        bfrag.y = Bc[(size_t)(k + 1) * DIM];
        acc = __builtin_amdgcn_wmma_f32_16x16x4_f32(
            false, a, false, bfrag, (short)0, acc, false, false);
    }
    #pragma unroll
    for (int j = 0; j < 8; ++j) {
        int m = tm * 16 + j + 8 * h;
        ws[WS_V + m * DIM + tn * 16 + r] = acc[j];
    }
}

// ------------------- pass 2: s[b,n] = (x[b,n].V[b] + c[b]) / (N*sqrt(D)) ---
// one wave per row; 8 waves per block; 8192 blocks
__global__ void scores_kernel(const float* __restrict__ x, float* __restrict__ ws) {
    const float invscale = 4.3158372875155487e-05f;  // 1/(1024*sqrt(512))
    int wave = threadIdx.x >> 5;
    int lane = threadIdx.x & 31;
    int row = blockIdx.x * 8 + wave;          // 0..65535
    int b = row >> 10;
    const float4* x4 = (const float4*)(x + (size_t)row * DIM);
    const float4* v4 = (const float4*)(ws + WS_V + (size_t)b * DIM);
    float s = 0.0f;
    #pragma unroll
    for (int i = 0; i < 4; ++i) {
        int idx = lane + 32 * i;              // coalesced across the wave
        float4 xv = x4[idx];
        float4 vv = v4[idx];
        s += xv.x * vv.x + xv.y * vv.y + xv.z * vv.z + xv.w * vv.w;
    }
    for (int off = 16; off > 0; off >>= 1)
        s += __shfl_down(s, off, 32);
    if (lane == 0)
        ws[WS_S + row] = (s + ws[WS_C + b]) * invscale;
}

// ------------------------------ softmax over n per batch -------------------
__global__ void softmax_kernel(const float* __restrict__ ws, float* __restrict__ attn_out) {
    __shared__ float red[256];
    int b = blockIdx.x;
    int t = threadIdx.x;
    const float* sb = ws + WS_S + b * SEQ;

    float vals[4];
    float m = -1e30f;
    #pragma unroll
    for (int i = 0; i < 4; ++i) {
        vals[i] = sb[t + 256 * i];
        m = fmaxf(m, vals[i]);
    }
    red[t] = m;
    __syncthreads();
    for (int stridev = 128; stridev > 0; stridev >>= 1) {
        if (t < stridev) red[t] = fmaxf(red[t], red[t + stridev]);
        __syncthreads();
    }
    float mx = red[0];
    __syncthreads();

    float sum = 0.0f;
    #pragma unroll
    for (int i = 0; i < 4; ++i) {
        vals[i] = expf(vals[i] - mx);
        sum += vals[i];
    }
    red[t] = sum;
    __syncthreads();
    for (int stridev = 128; stridev > 0; stridev >>= 1) {
        if (t < stridev) red[t] += red[t + stridev];
        __syncthreads();
    }
    float inv = 1.0f / red[0];
    #pragma unroll
    for (int i = 0; i < 4; ++i)
        attn_out[b * SEQ + t + 256 * i] = vals[i] * inv;
}

// --------------- pass 3: agg[b,d] = sum_n attn[b,n] * x[b,n,d] -------------
__global__ void aggregate_kernel(const float* __restrict__ x,
                                 const float* __restrict__ attn,
                                 float* __restrict__ agg) {
    int b = blockIdx.x >> 3;
    int chunk = blockIdx.x & 7;
    const float* xb = x + (size_t)b * SEQ * DIM + (size_t)chunk * 128 * DIM;
    const float* ab = attn + b * SEQ + chunk * 128;
    int d0 = threadIdx.x, d1 = threadIdx.x + 256;
    float s0 = 0.0f, s1 = 0.0f;
    #pragma unroll 4
    for (int n = 0; n < 128; ++n) {
        float w = ab[n];
        s0 += w * xb[n * DIM + d0];
        s1 += w * xb[n * DIM + d1];
    }
    atomicAdd(&agg[b * DIM + d0], s0);
    atomicAdd(&agg[b * DIM + d1], s1);
}

// ---------------------------------------------------------------------------
extern "C" void kernel_launch(void* const* d_in, const int* in_sizes, int n_in,
                              void* d_out, int out_size, void* d_ws, size_t ws_size,
                              hipStream_t stream) {
    const float* x  = (const float*)d_in[0];
    const float* Wq = (const float*)d_in[1];
    const float* bq = (const float*)d_in[2];
    const float* Wk = (const float*)d_in[3];
    const float* bk = (const float*)d_in[4];
    float* out = (float*)d_out;                 // [0,32768): aggregated, [32768,98304): attn
    float* ws  = (float*)d_ws;
    float* attn_out = out + BATCH * DIM;

    init_zero_kernel<<<128, 256, 0, stream>>>(ws, out);
    colsum_kernel<<<BATCH * 8, 256, 0, stream>>>(x, ws);
    gemm_wk_kernel<<<32, 128, 0, stream>>>(Wk, bk, ws);
    dotc_kernel<<<BATCH, 32, 0, stream>>>(bq, ws);
    gemm_wq_kernel<<<32, 128, 0, stream>>>(Wq, ws);
    scores_kernel<<<8192, 256, 0, stream>>>(x, ws);
    softmax_kernel<<<BATCH, 256, 0, stream>>>(ws, attn_out);
    aggregate_kernel<<<BATCH * 8, 256, 0, stream>>>(x, attn_out, out);
}